// ResBlock_88510686036612
// MI455X (gfx1250) — compile-verified
//
#include <hip/hip_runtime.h>
#include <cstdint>
#include <cstddef>

#ifndef __has_builtin
#define __has_builtin(x) 0
#endif

#if __has_builtin(__builtin_amdgcn_global_load_async_to_lds_b32)
#define ASYNC_LDS 1
#else
#define ASYNC_LDS 0
#endif

#if __has_builtin(__builtin_amdgcn_s_wait_asynccnt)
#define WAIT_ASYNC_BUILTIN 1
#else
#define WAIT_ASYNC_BUILTIN 0
#endif

namespace {
constexpr int Bn = 8, Cn = 64, Hn = 64, Wn = 64, On = 64;
constexpr int TH  = 16;        // spatial tile: 16x16
constexpr int OB  = 16;        // output channels per block
constexpr int CCk = 16;        // input channels staged per LDS chunk
constexpr int HL  = TH + 2;    // 18 (tile + 1-halo each side)
constexpr float RES_SCALE = 0.1f;
}  // namespace

typedef __attribute__((address_space(1))) int* gptr_i32;
typedef __attribute__((address_space(3))) int* lptr_i32;

template <bool FIRST>
__global__ __launch_bounds__(256)
void adder_layer(const float* __restrict__ in,     // [8,64,64,64]
                 const float* __restrict__ w,      // [64,64,3,3]
                 const float* __restrict__ resid,  // x for residual (layer 2 only)
                 float* __restrict__ out)          // [8,64,64,64]
{
    __shared__ float tile[CCk][HL][HL];

    const int tid = threadIdx.x;
    const int tx  = tid & (TH - 1);
    const int ty  = tid >> 4;
    const int bx  = blockIdx.x & 3;   // 64/16 = 4 tiles per row
    const int by  = blockIdx.x >> 2;
    const int ob  = blockIdx.y * OB;  // output-channel base
    const int b   = blockIdx.z;
    const int gx0 = bx * TH, gy0 = by * TH;

    float acc[OB];
#pragma unroll
    for (int i = 0; i < OB; ++i) acc[i] = 0.0f;

    const float* inB = in + (size_t)b * Cn * Hn * Wn;

#pragma unroll 1
    for (int c0 = 0; c0 < Cn; c0 += CCk) {
        // ---- stage CCk channels of the 18x18 halo tile into LDS ----
#pragma unroll 1
        for (int i = tid; i < CCk * HL * HL; i += 256) {
            const int cc = i / (HL * HL);
            const int r  = i - cc * (HL * HL);
            const int yy = r / HL;
            const int xx = r - yy * HL;
            const int sy = gy0 + yy - 1;
            const int sx = gx0 + xx - 1;
            if ((unsigned)sy < (unsigned)Hn && (unsigned)sx < (unsigned)Wn) {
                const float* gp = inB + ((size_t)(c0 + cc) * Hn + sy) * Wn + sx;
#if ASYNC_LDS
                // gfx1250 async global->LDS copy (tracked by ASYNCcnt)
                __builtin_amdgcn_global_load_async_to_lds_b32(
                    (gptr_i32)gp,
                    (lptr_i32)&tile[cc][yy][xx],
                    0, 0);
#else
                tile[cc][yy][xx] = *gp;
#endif
            } else {
                tile[cc][yy][xx] = 0.0f;  // zero padding
            }
        }
#if ASYNC_LDS
#if WAIT_ASYNC_BUILTIN
        __builtin_amdgcn_s_wait_asynccnt(0);
#else
        asm volatile("s_wait_asynccnt 0x0" ::: "memory");
#endif
#endif
        __syncthreads();

        // ---- accumulate L1 distances for this channel chunk ----
#pragma unroll 1
        for (int c = 0; c < CCk; ++c) {
            float px[9];
#pragma unroll
            for (int kh = 0; kh < 3; ++kh)
#pragma unroll
                for (int kw = 0; kw < 3; ++kw)
                    px[kh * 3 + kw] = tile[c][ty + kh][tx + kw];

#pragma unroll
            for (int oo = 0; oo < OB; ++oo) {
                // Uniform address (blockIdx + loop counters) -> scalar (SMEM) loads
                const float* wp = w + ((size_t)(ob + oo) * Cn + (c0 + c)) * 9;
#pragma unroll
                for (int j = 0; j < 9; ++j) {
                    acc[oo] -= __builtin_fabsf(px[j] - wp[j]);  // v_sub + v_add |d|
                }
            }
        }
        __syncthreads();  // protect LDS before next chunk overwrites it
    }

    // ---- epilogue: ReLU (layer 1) or scale + residual (layer 2) ----
    const int h  = gy0 + ty;
    const int x_ = gx0 + tx;
    const size_t base = (((size_t)b * On + ob) * Hn + h) * Wn + x_;
#pragma unroll
    for (int oo = 0; oo < OB; ++oo) {
        const size_t oidx = base + (size_t)oo * Hn * Wn;
        float v = acc[oo];
        if (FIRST) {
            v = v > 0.0f ? v : 0.0f;
        } else {
            v = v * RES_SCALE + resid[oidx];
        }
        out[oidx] = v;
    }
}

extern "C" void kernel_launch(void* const* d_in, const int* in_sizes, int n_in,
                              void* d_out, int out_size, void* d_ws, size_t ws_size,
                              hipStream_t stream) {
    const float* x  = (const float*)d_in[0];
    const float* W1 = (const float*)d_in[1];
    const float* W2 = (const float*)d_in[2];
    float* out = (float*)d_out;
    float* y1  = (float*)d_ws;  // intermediate: 8*64*64*64 floats = 8 MiB

    dim3 grid((Hn / TH) * (Wn / TH), On / OB, Bn);  // 16 x 4 x 8 = 512 blocks
    dim3 block(256);

    adder_layer<true ><<<grid, block, 0, stream>>>(x,  W1, nullptr, y1);
    adder_layer<false><<<grid, block, 0, stream>>>(y1, W2, x,       out);
}